// VectorQuantizer_63359357551207
// MI455X (gfx1250) — compile-verified
//
#include <hip/hip_runtime.h>
#include <hip/hip_bf16.h>

typedef float v2f __attribute__((ext_vector_type(2)));
typedef float v8f __attribute__((ext_vector_type(8)));

#define CB   32
#define CC   256
#define HH   32
#define WW   32
#define KK   1024
#define NN   (CB*HH*WW)          // 32768 tokens
#define NUMEL (CB*CC*HH*WW)      // 8388608
#define TOK  16                  // tokens per workgroup
#define AS_STRIDE 260            // padded A-tile row (bank-conflict-free f32x2 reads)
#define LS_STRIDE 1024           // logits row stride (union with A region -> 64KB LDS)

#if defined(__has_builtin)
#if __has_builtin(__builtin_amdgcn_wmma_f32_16x16x4_f32)
#define USE_WMMA_F32 1
#endif
#endif

// ---------------- kernel 0: ||e_k||^2, zero histogram + loss accumulator ----
__global__ __launch_bounds__(256) void vq_init(const float* __restrict__ emb,
                                               float* __restrict__ ee,
                                               float* __restrict__ counts,
                                               float* __restrict__ loss_sum) {
  int k = blockIdx.x * 256 + threadIdx.x;
  if (k < KK) {
    const float* e = emb + (size_t)k * CC;
    float s = 0.f;
    #pragma unroll 4
    for (int c = 0; c < CC; ++c) { float v = e[c]; s += v * v; }
    ee[k] = s;
    counts[k] = 0.f;
  }
  if (k == 0) *loss_sum = 0.f;
}

// ---------------- kernel 1: fused GEMM + gumbel softmax + argmax ------------
// grid = N/16 blocks of 256 threads (8 wave32).
__global__ __launch_bounds__(256)
void vq_main(const float* __restrict__ inp, const float* __restrict__ emb,
             const float* __restrict__ gum, const float* __restrict__ ee,
             float* __restrict__ enc_out, int* __restrict__ idx_out,
             float* __restrict__ encval_out, float* __restrict__ counts) {
  __shared__ float Sh[TOK * LS_STRIDE];   // 64 KB: A-tile region reused for logits

  const int tid = threadIdx.x;
  const int n0  = blockIdx.x * TOK;       // 16 consecutive tokens: same b,h (W=32)
  const int b   = n0 >> 10;
  const int hw0 = n0 & 1023;              // h*32 + w0

  // ---- stage A tile: Sh[token u][c] = inputs[b, c, h, w0+u] (NCHW gather)
  {
    const int u  = tid & 15;
    const int c0 = tid >> 4;              // 0..15
    const float* src = inp + (size_t)b * CC * (HH * WW) + hw0 + u;
    #pragma unroll
    for (int i = 0; i < 16; ++i) {
      int c = c0 + i * 16;
      Sh[u * AS_STRIDE + c] = src[(size_t)c * (HH * WW)];
    }
  }
  __syncthreads();

  const int wave = tid >> 5;
  const int lane = tid & 31;
  const int half = lane >> 4;             // VGPR half selects K/c pair
  const int lm   = lane & 15;
  const int kwave = wave * 128;           // this wave's 128 code columns

  v8f acc[8] = {};                        // 8 tiles of 16x16 f32 accumulators

  for (int c0 = 0; c0 < CC; c0 += 4) {
    // A operand (16x4 f32): lane lm = token row, c = c0 + 2*half + {0,1}
    v2f a = *(const v2f*)&Sh[lm * AS_STRIDE + c0 + half * 2];
    #pragma unroll
    for (int j = 0; j < 8; ++j) {
      int k = kwave + j * 16 + lm;        // B col = code index (B = emb^T tile)
      const float* ep = emb + (size_t)k * CC + c0 + half * 2;
#ifdef USE_WMMA_F32
      v2f bv = *(const v2f*)ep;
      acc[j] = __builtin_amdgcn_wmma_f32_16x16x4_f32(
          false, a, false, bv, (short)0, acc[j], false, false);
#else
      const float* ap = &Sh[0];           // scalar fallback (keeps build green)
      #pragma unroll
      for (int r = 0; r < 8; ++r) {
        int t = r + half * 8;
        const float* av = &ap[t * AS_STRIDE + c0];
        acc[j][r] += av[0]*ep[0] + av[1]*ep[1] + av[2]*ep[2] + av[3]*ep[3];
      }
#endif
    }
  }

  __syncthreads();                        // all waves done reading A region
  // ---- spill 2*dot to logits layout (C/D layout: VGPR r -> token r / r+8)
  #pragma unroll
  for (int j = 0; j < 8; ++j) {
    #pragma unroll
    for (int r = 0; r < 8; ++r) {
      int t = r + half * 8;
      int k = kwave + j * 16 + lm;
      Sh[t * LS_STRIDE + k] = 2.0f * acc[j][r];
    }
  }
  __syncthreads();

  // ---- per-row gumbel softmax + argmax: 16 threads per token row
  const int g = tid >> 4;                 // token row 0..15
  const int u = tid & 15;                 // lane within row group
  const int n = n0 + g;
  float* row = Sh + g * LS_STRIDE;
  const float* grow = gum + (size_t)n * KK;

  float m = -3.4e38f; int mi = 0;
  #pragma unroll 4
  for (int j = 0; j < 64; ++j) {
    int k = j * 16 + u;
    float v = row[k] + grow[k] - ee[k];   // logits + gumbel (row-const dropped)
    row[k] = v;
    if (v > m) { m = v; mi = k; }
  }
  #pragma unroll
  for (int off = 1; off < 16; off <<= 1) {
    float om = __shfl_xor(m, off, 32);
    int   oi = __shfl_xor(mi, off, 32);
    if (om > m || (om == m && oi < mi)) { m = om; mi = oi; }
  }

  float s = 0.f;
  #pragma unroll 4
  for (int j = 0; j < 64; ++j) s += expf(row[j * 16 + u] - m);
  #pragma unroll
  for (int off = 1; off < 16; off <<= 1) s += __shfl_xor(s, off, 32);
  const float invs = 1.0f / s;

  float* erow = enc_out + (size_t)n * KK;
  #pragma unroll 4
  for (int j = 0; j < 64; ++j) {
    int k = j * 16 + u;
    float y = expf(row[k] - m) * invs;
    float hard = (k == mi) ? 1.0f : 0.0f;
    erow[k] = (hard + y) - y;             // straight-through forward value
  }
  if (u == 0) {
    float ymax = invs;                    // exp(0)/sum
    float ev = (1.0f + ymax) - ymax;
    idx_out[n] = mi;
    encval_out[n] = ev;
    atomicAdd(&counts[mi], ev);
  }
}

// ---------------- kernel 2: gather-quantize, ST output, loss reduction ------
__global__ __launch_bounds__(256)
void vq_quant(const float* __restrict__ inp, const float* __restrict__ emb,
              const int* __restrict__ idx, const float* __restrict__ encval,
              float* __restrict__ qout, float* __restrict__ loss_sum) {
  const int flat = blockIdx.x * 256 + threadIdx.x;   // NCHW flat index
  const int c = (flat >> 10) & 255;
  const int b = flat >> 18;
  const int n = b * 1024 + (flat & 1023);            // token = (b, h*32+w)
  float x  = inp[flat];
  float q  = encval[n] * emb[(size_t)idx[n] * CC + c];
  float d  = q - x;
  qout[flat] = x + d;                                // x + (quantized - x)

  float v = d * d;
  __shared__ float red[8];
  #pragma unroll
  for (int off = 16; off >= 1; off >>= 1) v += __shfl_down(v, off, 32);
  if ((threadIdx.x & 31) == 0) red[threadIdx.x >> 5] = v;
  __syncthreads();
  if (threadIdx.x < 32) {
    float t = (threadIdx.x < 8) ? red[threadIdx.x] : 0.0f;
    #pragma unroll
    for (int off = 4; off >= 1; off >>= 1) t += __shfl_down(t, off, 32);
    if (threadIdx.x == 0) atomicAdd(loss_sum, t);
  }
}

// ---------------- kernel 3: scalars (loss, perplexity) ----------------------
__global__ __launch_bounds__(256)
void vq_final(const float* __restrict__ counts, const float* __restrict__ loss_sum,
              float* __restrict__ out_loss, float* __restrict__ out_perp) {
  float ent = 0.f;
  #pragma unroll
  for (int i = 0; i < 4; ++i) {
    int k = threadIdx.x + i * 256;
    float p = counts[k] * (1.0f / NN);
    ent += p * logf(p + 1e-10f);
  }
  __shared__ float red[8];
  #pragma unroll
  for (int off = 16; off >= 1; off >>= 1) ent += __shfl_down(ent, off, 32);
  if ((threadIdx.x & 31) == 0) red[threadIdx.x >> 5] = ent;
  __syncthreads();
  if (threadIdx.x == 0) {
    float e = 0.f;
    for (int i = 0; i < 8; ++i) e += red[i];
    *out_perp = expf(-e);
    *out_loss = 1.25f * (*loss_sum) * (1.0f / (float)NUMEL); // codebook + 0.25*commit
  }
}

extern "C" void kernel_launch(void* const* d_in, const int* in_sizes, int n_in,
                              void* d_out, int out_size, void* d_ws, size_t ws_size,
                              hipStream_t stream) {
  (void)in_sizes; (void)n_in; (void)out_size; (void)ws_size;
  const float* inp = (const float*)d_in[0];   // [32,256,32,32] f32
  const float* emb = (const float*)d_in[1];   // [1024,256] f32
  const float* gum = (const float*)d_in[2];   // [32768,1024] f32
  // d_in[3] = is_training (==1): training path implemented.

  float* out    = (float*)d_out;
  float* q_st   = out;                        // [32,256,32,32]
  float* loss_p = out + NUMEL;                // scalar
  float* perp_p = out + NUMEL + 1;            // scalar
  float* enc    = out + NUMEL + 2;            // [32768,1024]

  // workspace: ee(4KB) | counts(4KB) | idx(128KB) | encval(128KB) | loss(4B)
  char*  ws       = (char*)d_ws;
  float* ee       = (float*)ws;
  float* counts   = (float*)(ws + 4096);
  int*   idxb     = (int*)  (ws + 8192);
  float* encval   = (float*)(ws + 8192 + NN * sizeof(int));
  float* loss_sum = (float*)(ws + 8192 + NN * sizeof(int) + NN * sizeof(float));

  vq_init <<<KK / 256, 256, 0, stream>>>(emb, ee, counts, loss_sum);
  vq_main <<<NN / TOK, 256, 0, stream>>>(inp, emb, gum, ee, enc, idxb, encval, counts);
  vq_quant<<<NUMEL / 256, 256, 0, stream>>>(inp, emb, idxb, encval, q_st, loss_sum);
  vq_final<<<1, 256, 0, stream>>>(counts, loss_sum, loss_p, perp_p);
}